// PointNetFPModule_6176162972234
// MI455X (gfx1250) — compile-verified
//
#include <hip/hip_runtime.h>
#include <math.h>

typedef __attribute__((ext_vector_type(2))) float v2f;
typedef __attribute__((ext_vector_type(8))) float v8f;

#define BS     8
#define M_PTS  1024
#define N_PTS  4096
#define C0     256
#define C1     128
#define C2     256
#define CIN1   (C0 + C1)   // 384

// ---------------- Kernel 1: 3-NN + inverse-distance weights ----------------
// One block = 256 query points of one batch; reference cloud staged in LDS.
__global__ __launch_bounds__(256) void three_nn_kernel(
    const float* __restrict__ xyz,      // (bs, m, 3)
    const float* __restrict__ pxyz,     // (bs, n, 3)
    int*  __restrict__ idx_out,         // (bs, n, 3)
    float* __restrict__ w_out) {        // (bs, n, 3) normalized weights
  __shared__ float sx[M_PTS], sy[M_PTS], sz[M_PTS];
  const int b   = blockIdx.y;
  const int tid = threadIdx.x;
  const float* src = xyz + (size_t)b * M_PTS * 3;
  for (int i = tid; i < M_PTS; i += 256) {
    sx[i] = src[i * 3 + 0];
    sy[i] = src[i * 3 + 1];
    sz[i] = src[i * 3 + 2];
  }
  __syncthreads();

  const int j = blockIdx.x * 256 + tid;
  const float* q = pxyz + ((size_t)b * N_PTS + j) * 3;
  const float px = q[0], py = q[1], pz = q[2];

  float d0 = 3.4e38f, d1 = 3.4e38f, d2 = 3.4e38f;
  int   i0 = 0, i1 = 0, i2 = 0;
  for (int i = 0; i < M_PTS; ++i) {
    float dx = px - sx[i], dy = py - sy[i], dz = pz - sz[i];
    float d  = dx * dx + dy * dy + dz * dz;
    if (d < d0)      { d2 = d1; i2 = i1; d1 = d0; i1 = i0; d0 = d; i0 = i; }
    else if (d < d1) { d2 = d1; i2 = i1; d1 = d;  i1 = i; }
    else if (d < d2) { d2 = d;  i2 = i; }
  }
  float w0 = 1.f / (sqrtf(fmaxf(d0, 0.f)) + 1e-8f);
  float w1 = 1.f / (sqrtf(fmaxf(d1, 0.f)) + 1e-8f);
  float w2 = 1.f / (sqrtf(fmaxf(d2, 0.f)) + 1e-8f);
  float inv = 1.f / (w0 + w1 + w2);
  size_t o = ((size_t)b * N_PTS + j) * 3;
  idx_out[o + 0] = i0; idx_out[o + 1] = i1; idx_out[o + 2] = i2;
  w_out[o + 0] = w0 * inv; w_out[o + 1] = w1 * inv; w_out[o + 2] = w2 * inv;
}

// ---------------- Kernel 2: 3-point weighted gather (interpolation) --------
__global__ __launch_bounds__(256) void interp_kernel(
    const float* __restrict__ feats,    // (bs, c0, m)
    const int*   __restrict__ idx,      // (bs, n, 3)
    const float* __restrict__ w,        // (bs, n, 3)
    float* __restrict__ up) {           // (bs, c0, n)
  const int b = blockIdx.z;
  const int c = blockIdx.y;
  const int j = blockIdx.x * 256 + threadIdx.x;
  const size_t o = ((size_t)b * N_PTS + j) * 3;
  const int   i0 = idx[o + 0], i1 = idx[o + 1], i2 = idx[o + 2];
  const float w0 = w[o + 0],  w1 = w[o + 1],  w2 = w[o + 2];
  const float* f = feats + ((size_t)b * C0 + c) * M_PTS;
  up[((size_t)b * C0 + c) * N_PTS + j] = w0 * f[i0] + w1 * f[i1] + w2 * f[i2];
}

// ---------------- WMMA f32 GEMM: D = relu(W x B + bias) --------------------
// One wave computes a 16(M) x 64(N) tile: 4 accumulators, A reused 4x.
// Lane decomposition per ISA layout tables: lane = 16*half + lid.
//   A vgpr i  holds A[M = lid,        K = 2*half + i]
//   B vgpr i  holds B[K = 2*half + i, N = lid]
//   D vgpr r  holds D[M = r + 8*half, N = lid]
__device__ __forceinline__ void accum_panel(
    v8f acc[4],
    const float* __restrict__ Wrow,   // &W[(M0+lid)*ldw + kOff]
    const float* __restrict__ B,      // batch base + N0 (row stride N_PTS)
    int K, int lid, int half) {
  for (int k0 = 0; k0 < K; k0 += 4) {
    const float* wp = Wrow + k0 + 2 * half;
    v2f a;
    a.x = wp[0];
    a.y = wp[1];
    const float* bp = B + (size_t)(k0 + 2 * half) * N_PTS + lid;
#pragma unroll
    for (int t = 0; t < 4; ++t) {
      v2f bf;
      bf.x = bp[t * 16];
      bf.y = bp[t * 16 + N_PTS];
      acc[t] = __builtin_amdgcn_wmma_f32_16x16x4_f32(
          false, a, false, bf, (short)0, acc[t], false, false);
    }
  }
}

__global__ __launch_bounds__(256) void gemm_bias_relu_wmma(
    const float* __restrict__ W, int ldw,
    const float* __restrict__ B1, long sB1, int K1,
    const float* __restrict__ B2, long sB2, int K2, int kOff2,
    const float* __restrict__ bias,
    float* __restrict__ out, long sOut) {
  const int lane = threadIdx.x & 31;
  const int wave = threadIdx.x >> 5;
  const int lid  = lane & 15;
  const int half = lane >> 4;

  const int MT = C2 / 16;            // 16 M-tiles
  const int NT = N_PTS / 64;         // 64 N-supertiles
  const int wid = blockIdx.x * 8 + wave;
  const int b   = wid / (MT * NT);
  const int rem = wid % (MT * NT);
  const int M0  = (rem / NT) * 16;
  const int N0  = (rem % NT) * 64;

  v8f acc[4];
#pragma unroll
  for (int t = 0; t < 4; ++t) acc[t] = (v8f){};

  const float* Wrow = W + (size_t)(M0 + lid) * ldw;
  accum_panel(acc, Wrow, B1 + (size_t)b * sB1 + N0, K1, lid, half);
  if (K2 > 0)   // uniform branch (scalar arg) — no EXEC divergence
    accum_panel(acc, Wrow + kOff2, B2 + (size_t)b * sB2 + N0, K2, lid, half);

  float* o = out + (size_t)b * sOut;
#pragma unroll
  for (int t = 0; t < 4; ++t) {
#pragma unroll
    for (int r = 0; r < 8; ++r) {
      const int Mrow = M0 + r + 8 * half;
      const float v = acc[t][r] + bias[Mrow];
      o[(size_t)Mrow * N_PTS + N0 + t * 16 + lid] = fmaxf(v, 0.f);
    }
  }
}

// ---------------------------------------------------------------------------
extern "C" void kernel_launch(void* const* d_in, const int* in_sizes, int n_in,
                              void* d_out, int out_size, void* d_ws, size_t ws_size,
                              hipStream_t stream) {
  (void)in_sizes; (void)n_in; (void)out_size; (void)ws_size;
  const float* xyz   = (const float*)d_in[0];   // (8,1024,3)
  const float* pxyz  = (const float*)d_in[1];   // (8,4096,3)
  const float* feats = (const float*)d_in[2];   // (8,256,1024)
  const float* skip  = (const float*)d_in[3];   // (8,128,4096)
  const float* W1    = (const float*)d_in[4];   // (256,384)
  const float* b1    = (const float*)d_in[5];   // (256,)
  const float* W2    = (const float*)d_in[6];   // (256,256)
  const float* b2    = (const float*)d_in[7];   // (256,)
  float* out = (float*)d_out;                   // (8,256,4096)

  char* ws = (char*)d_ws;
  size_t off = 0;
  int*   idx = (int*)(ws + off);   off += (size_t)BS * N_PTS * 3 * sizeof(int);
  float* wgt = (float*)(ws + off); off += (size_t)BS * N_PTS * 3 * sizeof(float);
  float* up  = (float*)(ws + off); off += (size_t)BS * C0 * N_PTS * sizeof(float);
  float* H   = (float*)(ws + off);

  three_nn_kernel<<<dim3(N_PTS / 256, BS), 256, 0, stream>>>(xyz, pxyz, idx, wgt);
  interp_kernel<<<dim3(N_PTS / 256, C0, BS), 256, 0, stream>>>(feats, idx, wgt, up);

  const int wavesTotal = BS * (C2 / 16) * (N_PTS / 64);   // 8192 waves
  // GEMM1: H = relu(W1 x [up; skip] + b1)  (virtual concat: two K-passes)
  gemm_bias_relu_wmma<<<dim3(wavesTotal / 8), 256, 0, stream>>>(
      W1, CIN1,
      up,   (long)C0 * N_PTS, C0,
      skip, (long)C1 * N_PTS, C1, C0,
      b1, H, (long)C2 * N_PTS);
  // GEMM2: out = relu(W2 x H + b2)
  gemm_bias_relu_wmma<<<dim3(wavesTotal / 8), 256, 0, stream>>>(
      W2, C2,
      H, (long)C2 * N_PTS, C2,
      nullptr, 0, 0, 0,
      b2, out, (long)C2 * N_PTS);
}